// MambaBlock_67834713473024
// MI455X (gfx1250) — compile-verified
//
#include <hip/hip_runtime.h>
#include <hip/hip_bf16.h>

// Mamba block for gfx1250 (MI455X), fp32 end-to-end.
// GEMMs: V_WMMA_F32_16X16X4_F32, double-buffered LDS fed by CDNA5 async DMA
//        (global_load_async_to_lds_b128, ASYNCcnt-tracked) -> no VGPR staging.
// Scan:  3-pass chunked linear scan (64 chunks x 64 steps) exploiting
//        prod(exp(dt*A)) = exp(A*sum(dt)) -> 64x more parallelism.

typedef float v2f __attribute__((ext_vector_type(2)));
typedef float v8f __attribute__((ext_vector_type(8)));

#define BM 128
#define BN 64
#define BK 16

#define SEQ   4096
#define DI    2048   // D_INNER
#define NC    64     // chunks
#define CL    64     // chunk length

// C(M x N) = A(M x K, row-major) * W(N x K, row-major)^T
// M % 128 == 0, N % 64 == 0, K % 16 == 0.
//
// LDS layout (group-major, 16B-aligned cells for B128 async DMA):
//   As[buf][q][m][j] = A[m][4q+j]  -> frag read addr = q*2048 + m*16 + 8*lh (+4)
//   bank(m) = (4m + 2lh) mod 64    -> conflict-free across 16 lanes, halves disjoint
__global__ __launch_bounds__(256)
void wmma_gemm_f32(const float* __restrict__ A, const float* __restrict__ W,
                   float* __restrict__ C, int M, int N, int K) {
    __shared__ float As[2][4][BM][4];
    __shared__ float Bs[2][4][BN][4];

    const int tid  = threadIdx.x;
    const int wave = tid >> 5;        // 0..7
    const int lane = tid & 31;
    const int lh   = lane >> 4;       // lane-half: selects K pair within group
    const int lm   = lane & 15;
    const int m0   = blockIdx.y * BM;
    const int n0   = blockIdx.x * BN;
    const int wm   = wave * 16;       // wave's 16-row strip within block tile

    // Per-thread DMA coordinates (A: 8 floats/thread, B: 4 floats/thread)
    const int am = (tid * 8) >> 4;    // A tile row; two threads per row
    const int ak = (tid * 8) & 15;    // 0 or 8  -> groups ak/4, ak/4+1
    const int bn = (tid * 4) >> 4;    // B tile row
    const int bk = (tid * 4) & 15;    // 0,4,8,12 -> group bk/4

    const float* Ag = A + (size_t)(m0 + am) * K + ak;
    const float* Wg = W + (size_t)(n0 + bn) * K + bk;

    // LDS destination byte addresses (low 32 bits of generic addr == LDS offset)
    unsigned ldsA0[2], ldsA1[2], ldsB0[2];
#pragma unroll
    for (int b = 0; b < 2; ++b) {
        ldsA0[b] = (unsigned)(size_t)&As[b][(ak >> 2) + 0][am][0];
        ldsA1[b] = (unsigned)(size_t)&As[b][(ak >> 2) + 1][am][0];
        ldsB0[b] = (unsigned)(size_t)&Bs[b][(bk >> 2) + 0][bn][0];
    }

    v8f acc[4];
#pragma unroll
    for (int i = 0; i < 4; ++i)
#pragma unroll
        for (int e = 0; e < 8; ++e) acc[i][e] = 0.0f;

    // Prologue: async-DMA K-tile 0 into buffer 0.
    {
        const float* a0 = Ag;
        const float* a1 = Ag + 4;
        const float* b0 = Wg;
        asm volatile("global_load_async_to_lds_b128 %0, %1, off"
                     :: "v"(ldsA0[0]), "v"(a0) : "memory");
        asm volatile("global_load_async_to_lds_b128 %0, %1, off"
                     :: "v"(ldsA1[0]), "v"(a1) : "memory");
        asm volatile("global_load_async_to_lds_b128 %0, %1, off"
                     :: "v"(ldsB0[0]), "v"(b0) : "memory");
    }

    const int nk = K / BK;
    for (int i = 0; i < nk; ++i) {
        // Own DMA landed, then all waves' DMA landed.
        asm volatile("s_wait_asynccnt 0x0" ::: "memory");
        __syncthreads();

        // Kick off next tile's DMA into the other buffer; overlaps the WMMAs.
        if (i + 1 < nk) {
            const int nb = (i + 1) & 1;
            const float* a0 = Ag + (i + 1) * BK;
            const float* a1 = Ag + (i + 1) * BK + 4;
            const float* b0 = Wg + (i + 1) * BK;
            asm volatile("global_load_async_to_lds_b128 %0, %1, off"
                         :: "v"(ldsA0[nb]), "v"(a0) : "memory");
            asm volatile("global_load_async_to_lds_b128 %0, %1, off"
                         :: "v"(ldsA1[nb]), "v"(a1) : "memory");
            asm volatile("global_load_async_to_lds_b128 %0, %1, off"
                         :: "v"(ldsB0[nb]), "v"(b0) : "memory");
        }

        const int buf = i & 1;
        // Load ALL fragments first, then issue 16 WMMAs back-to-back.
        float af[4][2];
        float bf[4][4][2];
#pragma unroll
        for (int q = 0; q < 4; ++q) {
            af[q][0] = As[buf][q][wm + lm][2 * lh + 0];
            af[q][1] = As[buf][q][wm + lm][2 * lh + 1];
#pragma unroll
            for (int nt = 0; nt < 4; ++nt) {
                bf[q][nt][0] = Bs[buf][q][nt * 16 + lm][2 * lh + 0];
                bf[q][nt][1] = Bs[buf][q][nt * 16 + lm][2 * lh + 1];
            }
        }
#pragma unroll
        for (int q = 0; q < 4; ++q) {
            v2f a; a[0] = af[q][0]; a[1] = af[q][1];
#pragma unroll
            for (int nt = 0; nt < 4; ++nt) {
                v2f b; b[0] = bf[q][nt][0]; b[1] = bf[q][nt][1];
                acc[nt] = __builtin_amdgcn_wmma_f32_16x16x4_f32(
                    false, a, false, b, (short)0, acc[nt], false, false);
            }
        }
    }

    // C/D layout: VGPR r, lanes 0-15 -> row r, lanes 16-31 -> row r+8; col = lm.
#pragma unroll
    for (int nt = 0; nt < 4; ++nt)
#pragma unroll
        for (int r = 0; r < 8; ++r)
            C[(size_t)(m0 + wm + r + 8 * lh) * N + n0 + nt * 16 + lm] = acc[nt][r];
}

// Depthwise causal conv(4) + bias + SiLU. xi = xz[:, 0:2048] -> xc (4096,2048).
__global__ __launch_bounds__(256)
void conv_silu_kernel(const float* __restrict__ xz, const float* __restrict__ w,
                      const float* __restrict__ b, float* __restrict__ xc) {
    int idx = blockIdx.x * blockDim.x + threadIdx.x;   // SEQ*DI
    int d = idx & (DI - 1);
    int t = idx >> 11;
    float acc = b[d];
#pragma unroll
    for (int j = 0; j < 4; ++j) {
        int tt = t - 3 + j;
        if (tt >= 0) acc += w[d * 4 + j] * xz[tt * 4096 + d];
    }
    xc[idx] = acc / (1.0f + __expf(-acc));
}

// Zero-pad x_proj_w (33 x 2048) -> (64 x 2048) so the WMMA GEMM applies.
__global__ __launch_bounds__(256)
void pad_w_kernel(const float* __restrict__ xw, float* __restrict__ wpad) {
    int idx = blockIdx.x * blockDim.x + threadIdx.x;   // 64*2048
    int n = idx >> 11;
    int k = idx & (DI - 1);
    wpad[idx] = (n < 33) ? xw[n * 2048 + k] : 0.0f;
}

__device__ __forceinline__ float softplus_f(float x) {
    return (x > 20.0f) ? x : __logf(1.0f + __expf(x));
}

// ---- Chunked selective scan ----
// xp (4096 x 64 padded): col 0 = dt_raw, 1..16 = B, 17..32 = C.
//
// Pass 1: per (chunk c, channel d): local scan from h=0 over CL steps.
__global__ __launch_bounds__(256)
void scan_pass1(const float* __restrict__ xp, const float* __restrict__ xc,
                const float* __restrict__ dt_w, const float* __restrict__ dt_b,
                const float* __restrict__ A_log,
                float* __restrict__ hloc, float* __restrict__ sumdt) {
    const int d = blockIdx.x * blockDim.x + threadIdx.x;  // 0..DI-1
    const int c = blockIdx.y;                              // 0..NC-1
    float An[16];
#pragma unroll
    for (int n = 0; n < 16; ++n) An[n] = -__expf(A_log[d * 16 + n]);
    const float dtw = dt_w[d];
    const float dtb = dt_b[d];
    float h[16];
#pragma unroll
    for (int n = 0; n < 16; ++n) h[n] = 0.0f;
    float sd = 0.0f;

    const int t0 = c * CL;
    for (int tt = 0; tt < CL; ++tt) {
        const int t = t0 + tt;
        const float* xpr = xp + t * 64;        // uniform within block
        const float dt = softplus_f(xpr[0] * dtw + dtb);
        sd += dt;
        const float dtx = dt * xc[t * DI + d];
#pragma unroll
        for (int n = 0; n < 16; ++n)
            h[n] = __expf(dt * An[n]) * h[n] + dtx * xpr[1 + n];
    }
    sumdt[c * DI + d] = sd;
#pragma unroll
    for (int n = 0; n < 16; ++n) hloc[(size_t)(c * DI + d) * 16 + n] = h[n];
}

// Pass 2: sequential carry over the 64 chunks, parallel over all (d,n).
// prod of dA over a chunk = exp(A[d,n] * sumdt[c,d]). In-place hloc -> hstart.
__global__ __launch_bounds__(256)
void scan_pass2(const float* __restrict__ A_log, const float* __restrict__ sumdt,
                float* __restrict__ hbuf) {
    const int id = blockIdx.x * blockDim.x + threadIdx.x;  // 0..DI*16-1
    const int d = id >> 4;
    const float A = -__expf(A_log[id]);
    float h = 0.0f;
    for (int c = 0; c < NC; ++c) {
        const float P = __expf(A * sumdt[c * DI + d]);
        const float hl = hbuf[(size_t)c * DI * 16 + id];
        hbuf[(size_t)c * DI * 16 + id] = h;   // chunk-entry state
        h = P * h + hl;
    }
}

// Pass 3: replay each chunk from its correct entry state; emit
// yact[t,d] = (h.C + xc*D) * silu(z).
__global__ __launch_bounds__(256)
void scan_pass3(const float* __restrict__ xp, const float* __restrict__ xc,
                const float* __restrict__ xz, const float* __restrict__ dt_w,
                const float* __restrict__ dt_b, const float* __restrict__ A_log,
                const float* __restrict__ Dp, const float* __restrict__ hstart,
                float* __restrict__ yact) {
    const int d = blockIdx.x * blockDim.x + threadIdx.x;  // 0..DI-1
    const int c = blockIdx.y;                              // 0..NC-1
    float An[16];
#pragma unroll
    for (int n = 0; n < 16; ++n) An[n] = -__expf(A_log[d * 16 + n]);
    const float dtw = dt_w[d];
    const float dtb = dt_b[d];
    const float Dd  = Dp[d];
    float h[16];
#pragma unroll
    for (int n = 0; n < 16; ++n) h[n] = hstart[(size_t)(c * DI + d) * 16 + n];

    const int t0 = c * CL;
    for (int tt = 0; tt < CL; ++tt) {
        const int t = t0 + tt;
        const float* xpr = xp + t * 64;
        const float dt  = softplus_f(xpr[0] * dtw + dtb);
        const float xcv = xc[t * DI + d];
        const float zz  = xz[t * 4096 + 2048 + d];
        const float sz  = zz / (1.0f + __expf(-zz));
        const float dtx = dt * xcv;
        float y = 0.0f;
#pragma unroll
        for (int n = 0; n < 16; ++n) {
            const float hn = __expf(dt * An[n]) * h[n] + dtx * xpr[1 + n];
            h[n] = hn;
            y += hn * xpr[17 + n];
        }
        yact[t * DI + d] = (y + xcv * Dd) * sz;
    }
}

extern "C" void kernel_launch(void* const* d_in, const int* in_sizes, int n_in,
                              void* d_out, int out_size, void* d_ws, size_t ws_size,
                              hipStream_t stream) {
    const float* x          = (const float*)d_in[0];  // (1,4096,1024)
    const float* in_proj_w  = (const float*)d_in[1];  // (4096,1024)
    const float* conv_w     = (const float*)d_in[2];  // (2048,1,4)
    const float* conv_b     = (const float*)d_in[3];  // (2048,)
    const float* x_proj_w   = (const float*)d_in[4];  // (33,2048)
    const float* dt_proj_w  = (const float*)d_in[5];  // (2048,1)
    const float* dt_proj_b  = (const float*)d_in[6];  // (2048,)
    const float* A_log      = (const float*)d_in[7];  // (2048,16)
    const float* Dp         = (const float*)d_in[8];  // (2048,)
    const float* out_proj_w = (const float*)d_in[9];  // (1024,2048)
    float* out = (float*)d_out;                       // (1,4096,1024)

    char* ws = (char*)d_ws;
    float* xz    = (float*)(ws);                        // 4096*4096 = 64 MB
    float* xc    = (float*)(ws + (64ull  << 20));       // 4096*2048 = 32 MB
    float* yact  = (float*)(ws + (96ull  << 20));       // 4096*2048 = 32 MB
    float* xp    = (float*)(ws + (128ull << 20));       // 4096*64   =  1 MB
    float* wpad  = (float*)(ws + (129ull << 20));       // 64*2048   = .5 MB
    float* sumdt = (float*)(ws + (130ull << 20));       // 64*2048   = .5 MB
    float* hbuf  = (float*)(ws + (131ull << 20));       // 64*2048*16 = 8 MB

    // 1) xz = x @ in_proj_w^T : (4096 x 4096), K=1024
    wmma_gemm_f32<<<dim3(4096 / BN, 4096 / BM), 256, 0, stream>>>(
        x, in_proj_w, xz, 4096, 4096, 1024);

    // 2) xc = silu(causal_conv4(xi) + b)
    conv_silu_kernel<<<(SEQ * DI) / 256, 256, 0, stream>>>(xz, conv_w, conv_b, xc);

    // 3) xp = xc @ pad(x_proj_w)^T : (4096 x 64), K=2048
    pad_w_kernel<<<(64 * DI) / 256, 256, 0, stream>>>(x_proj_w, wpad);
    wmma_gemm_f32<<<dim3(64 / BN, 4096 / BM), 256, 0, stream>>>(
        xc, wpad, xp, 4096, 64, 2048);

    // 4) chunked selective scan (3 passes) + x*D + silu(z) gating -> yact
    scan_pass1<<<dim3(DI / 256, NC), 256, 0, stream>>>(
        xp, xc, dt_proj_w, dt_proj_b, A_log, hbuf, sumdt);
    scan_pass2<<<(DI * 16) / 256, 256, 0, stream>>>(A_log, sumdt, hbuf);
    scan_pass3<<<dim3(DI / 256, NC), 256, 0, stream>>>(
        xp, xc, xz, dt_proj_w, dt_proj_b, A_log, Dp, hbuf, yact);

    // 5) out = yact @ out_proj_w^T : (4096 x 1024), K=2048
    wmma_gemm_f32<<<dim3(1024 / BN, 4096 / BM), 256, 0, stream>>>(
        yact, out_proj_w, out, 4096, 1024, 2048);
}